// GAT_10754598109970
// MI455X (gfx1250) — compile-verified
//
#include <hip/hip_runtime.h>

typedef float v2f __attribute__((ext_vector_type(2)));
typedef float v8f __attribute__((ext_vector_type(8)));

constexpr int Hh  = 2;     // heads
constexpr int Cc  = 64;    // channels per head
constexpr int HC  = 128;   // H*C
constexpr int ED  = 8;     // edge feature dim
constexpr int Ll  = 3;     // layers
constexpr int MH  = 64;    // mlp hidden
constexpr int MO  = 6;     // mlp out

// ---- monotone float<->uint encoding for atomicMax on floats ----
__device__ __forceinline__ unsigned fenc(float f) {
  unsigned u = __float_as_uint(f);
  return (u & 0x80000000u) ? ~u : (u | 0x80000000u);
}
__device__ __forceinline__ float fdec(unsigned e) {
  unsigned u = (e & 0x80000000u) ? (e & 0x7FFFFFFFu) : ~e;
  return __uint_as_float(u);
}

// ---- precompute we[l][h][d] = sum_c W_edge[l][d][h*C+c] * att_edge[l][h][c] ----
__global__ void k_we(const float* __restrict__ W_edge, const float* __restrict__ att_edge,
                     float* __restrict__ we) {
  int t = threadIdx.x;
  if (t >= Ll * Hh * ED) return;
  int l = t / (Hh * ED);
  int r = t % (Hh * ED);
  int h = r / ED, j = r % ED;
  float s = 0.f;
  for (int c = 0; c < Cc; ++c)
    s += W_edge[((size_t)l * ED + j) * HC + h * Cc + c] * att_edge[((size_t)l * Hh + h) * Cc + c];
  we[t] = s;  // layout [l][h][j]
}

// ---- degree + loop-attr accumulation ----
__global__ void k_deg_loop(const long long* __restrict__ ei, const float* __restrict__ edge_attr,
                           float* __restrict__ deg, float* __restrict__ lsum, int E) {
  int e = blockIdx.x * blockDim.x + threadIdx.x;
  if (e >= E) return;
  int d = (int)ei[(size_t)E + e];
  atomicAdd(&deg[d], 1.0f);
  const float* ea = edge_attr + (size_t)e * ED;
  float* lp = lsum + (size_t)d * ED;
#pragma unroll
  for (int j = 0; j < ED; ++j) atomicAdd(&lp[j], ea[j]);
}

__global__ void k_loop_div(float* __restrict__ lsum, const float* __restrict__ deg, int total) {
  int i = blockIdx.x * blockDim.x + threadIdx.x;
  if (i >= total) return;
  lsum[i] /= fmaxf(deg[i / ED], 1.0f);
}

// ---- fp32 WMMA GEMM: D[M x NCOLS] = A[M x K] @ B[K x NCOLS], row-major.
// Compile-time K/NCOLS; each wave computes a 16 x (16*CT) output tile.
template <int K, int NCOLS, int CT>
__global__ void __launch_bounds__(256)
k_gemm_wmma(const float* __restrict__ A, const float* __restrict__ B,
            float* __restrict__ D, int M) {
  constexpr int CGROUPS = NCOLS / (16 * CT);
  int wid  = (blockIdx.x * blockDim.x + threadIdx.x) >> 5;  // wave id (wave32)
  int lane = threadIdx.x & 31;
  int rtiles = (M + 15) >> 4;
  if (wid >= rtiles * CGROUPS) return;          // wave-uniform
  int rt = wid / CGROUPS, cg = wid % CGROUPS;
  int r0 = rt << 4, c0 = cg * (16 * CT);

  // A 16x4 frag: lanes 0-15 -> (M=lane, K=kb..kb+1) with kb=0; lanes 16-31 -> kb=2
  int kb   = (lane >> 4) << 1;
  int arow = r0 + (lane & 15);
  if (arow >= M) arow = M - 1;                  // clamp: garbage rows masked at store
  const float* Ap = A + (size_t)arow * K + kb;
  // B 4x16 frag: lanes 0-15 -> (K=kb, N=lane); lanes 16-31 -> (K=kb+2, N=lane-16)
  const float* Bp = B + (size_t)kb * NCOLS + c0 + (lane & 15);

  v8f acc[CT];
#pragma unroll
  for (int t = 0; t < CT; ++t) acc[t] = {};

#pragma unroll
  for (int k0 = 0; k0 < K; k0 += 4) {
    v2f a = *reinterpret_cast<const v2f*>(Ap + k0);       // global_load_b64
#pragma unroll
    for (int t = 0; t < CT; ++t) {
      v2f b;
      b.x = Bp[(size_t)k0 * NCOLS + t * 16];              // row kb+k0
      b.y = Bp[(size_t)(k0 + 1) * NCOLS + t * 16];        // row kb+k0+1
      acc[t] = __builtin_amdgcn_wmma_f32_16x16x4_f32(false, a, false, b,
                                                     (short)0, acc[t], false, false);
    }
  }

  // C/D 16x16: VGPR v -> rows r0+v (lanes 0-15) / r0+v+8 (lanes 16-31)
  int srow = r0 + ((lane >> 4) << 3);
  int scol = c0 + (lane & 15);
#pragma unroll
  for (int t = 0; t < CT; ++t)
#pragma unroll
    for (int v = 0; v < 8; ++v) {
      int rr = srow + v;
      if (rr < M) D[(size_t)rr * NCOLS + scol + t * 16] = acc[t][v];
    }
}

// ---- per-node attention scalars: a_s[n,h], a_d[n,h] (one wave per node) ----
__global__ void k_asd(const float* __restrict__ xh, const float* __restrict__ att_s,
                      const float* __restrict__ att_d, float* __restrict__ a_src,
                      float* __restrict__ a_dst, int N) {
  int t = blockIdx.x * blockDim.x + threadIdx.x;
  int n = t >> 5, lane = t & 31;
  if (n >= N) return;
  const float* row = xh + (size_t)n * HC;
  float x0 = row[lane], x1 = row[lane + 32], x2 = row[lane + 64], x3 = row[lane + 96];
  float s0 = x0 * att_s[lane]      + x1 * att_s[lane + 32];
  float s1 = x2 * att_s[lane + 64] + x3 * att_s[lane + 96];
  float d0 = x0 * att_d[lane]      + x1 * att_d[lane + 32];
  float d1 = x2 * att_d[lane + 64] + x3 * att_d[lane + 96];
  for (int off = 16; off; off >>= 1) {
    s0 += __shfl_down(s0, off, 32);
    s1 += __shfl_down(s1, off, 32);
    d0 += __shfl_down(d0, off, 32);
    d1 += __shfl_down(d1, off, 32);
  }
  if (lane == 0) {
    a_src[n * 2] = s0; a_src[n * 2 + 1] = s1;
    a_dst[n * 2] = d0; a_dst[n * 2 + 1] = d1;
  }
}

// ---- edge pass 1: alpha = leaky(a_s[src]+a_d[dst]+a_e, 0.2); atomicMax per dst ----
__global__ void k_alpha_max(const long long* __restrict__ ei, const float* __restrict__ edge_attr,
                            const float* __restrict__ loop_attr, const float* __restrict__ we_l,
                            const float* __restrict__ a_src, const float* __restrict__ a_dst,
                            float* __restrict__ alpha, unsigned* __restrict__ amax,
                            int* __restrict__ s32, int* __restrict__ d32, int E, int EF) {
  int e = blockIdx.x * blockDim.x + threadIdx.x;
  if (e >= EF) return;
  int s, d; const float* ea;
  if (e < E) { s = (int)ei[e]; d = (int)ei[(size_t)E + e]; ea = edge_attr + (size_t)e * ED; }
  else       { s = d = e - E;  ea = loop_attr + (size_t)(e - E) * ED; }
  s32[e] = s; d32[e] = d;
  float ae0 = 0.f, ae1 = 0.f;
#pragma unroll
  for (int j = 0; j < ED; ++j) { float v = ea[j]; ae0 += v * we_l[j]; ae1 += v * we_l[ED + j]; }
  float al0 = a_src[s * 2]     + a_dst[d * 2]     + ae0; al0 = al0 > 0.f ? al0 : 0.2f * al0;
  float al1 = a_src[s * 2 + 1] + a_dst[d * 2 + 1] + ae1; al1 = al1 > 0.f ? al1 : 0.2f * al1;
  alpha[(size_t)e * 2]     = al0;
  alpha[(size_t)e * 2 + 1] = al1;
  atomicMax(&amax[d * 2],     fenc(al0));
  atomicMax(&amax[d * 2 + 1], fenc(al1));
}

// ---- edge pass 2: ex = exp(alpha - max[dst]); denom[dst] += ex ----
__global__ void k_exp_denom(const int* __restrict__ d32, const unsigned* __restrict__ amax,
                            float* __restrict__ alpha, float* __restrict__ denom, int EF) {
  int e = blockIdx.x * blockDim.x + threadIdx.x;
  if (e >= EF) return;
  int d = d32[e];
#pragma unroll
  for (int h = 0; h < Hh; ++h) {
    float m  = fdec(amax[d * 2 + h]);
    float ex = __expf(alpha[(size_t)e * 2 + h] - m);
    alpha[(size_t)e * 2 + h] = ex;
    atomicAdd(&denom[d * 2 + h], ex);
  }
}

// ---- edge pass 3: out[dst] += xh[src] * (ex/denom[dst]); one wave per edge ----
__global__ void k_scatter(const int* __restrict__ s32, const int* __restrict__ d32,
                          const float* __restrict__ alpha, const float* __restrict__ denom,
                          const float* __restrict__ xh, float* __restrict__ outb, int EF) {
  int t = blockIdx.x * blockDim.x + threadIdx.x;
  int e = t >> 5, lane = t & 31;
  if (e >= EF) return;
  int s = s32[e], d = d32[e];
  int h = lane >> 4;  // lanes 0-15 -> head 0 (c<64), lanes 16-31 -> head 1
  float coef = alpha[(size_t)e * 2 + h] / denom[d * 2 + h];
  const float4 v = *reinterpret_cast<const float4*>(xh + (size_t)s * HC + lane * 4);
  float* op = outb + (size_t)d * HC + lane * 4;
  atomicAdd(op + 0, v.x * coef);
  atomicAdd(op + 1, v.y * coef);
  atomicAdd(op + 2, v.z * coef);
  atomicAdd(op + 3, v.w * coef);
}

// ---- bias + leaky-relu (in place) ----
__global__ void k_bias_leaky(float* __restrict__ buf, const float* __restrict__ bias,
                             long total, int ncols, float slope) {
  long i = blockIdx.x * (long)blockDim.x + threadIdx.x;
  if (i >= total) return;
  float v = buf[i] + bias[i % ncols];
  buf[i] = v > 0.f ? v : slope * v;
}

// ---- final MLP layer: out = hidden @ w2 + b2 ----
__global__ void k_mlp2(const float* __restrict__ hid, const float* __restrict__ w2,
                       const float* __restrict__ b2, float* __restrict__ out, int N) {
  __shared__ float sw[MH * MO + MO];
  for (int i = threadIdx.x; i < MH * MO + MO; i += blockDim.x)
    sw[i] = (i < MH * MO) ? w2[i] : b2[i - MH * MO];
  __syncthreads();
  int n = blockIdx.x * blockDim.x + threadIdx.x;
  if (n >= N) return;
  const float* hr = hid + (size_t)n * MH;
  float acc[MO];
#pragma unroll
  for (int o = 0; o < MO; ++o) acc[o] = sw[MH * MO + o];
  for (int j = 0; j < MH; ++j) {
    float hv = hr[j];
#pragma unroll
    for (int o = 0; o < MO; ++o) acc[o] += hv * sw[j * MO + o];
  }
#pragma unroll
  for (int o = 0; o < MO; ++o) out[(size_t)n * MO + o] = acc[o];
}

static inline int gridFor(long n, int b) { return (int)((n + b - 1) / b); }

extern "C" void kernel_launch(void* const* d_in, const int* in_sizes, int n_in,
                              void* d_out, int out_size, void* d_ws, size_t ws_size,
                              hipStream_t stream) {
  const float*     x         = (const float*)d_in[0];
  const long long* ei        = (const long long*)d_in[1];
  const float*     edge_attr = (const float*)d_in[2];
  const float*     W0        = (const float*)d_in[3];
  const float*     W12       = (const float*)d_in[4];
  const float*     att_src   = (const float*)d_in[5];
  const float*     att_dst   = (const float*)d_in[6];
  const float*     att_edge  = (const float*)d_in[7];
  const float*     W_edge    = (const float*)d_in[8];
  const float*     bias      = (const float*)d_in[9];
  const float*     mlp_w1    = (const float*)d_in[10];
  const float*     mlp_b1    = (const float*)d_in[11];
  const float*     mlp_w2    = (const float*)d_in[12];
  const float*     mlp_b2    = (const float*)d_in[13];

  int N  = in_sizes[0] / 16;
  int E  = in_sizes[1] / 2;
  int EF = E + N;

  // ---- workspace carving (256B aligned) ----
  char* ws = (char*)d_ws;
  size_t off = 0;
  auto take = [&](size_t bytes) -> char* {
    char* p = ws + off;
    off = (off + bytes + 255) & ~(size_t)255;
    return p;
  };
  float*    we    = (float*)take((size_t)Ll * Hh * ED * 4);
  float*    deg   = (float*)take((size_t)N * 4);
  float*    lsum  = (float*)take((size_t)N * ED * 4);      // becomes loop_attr
  float*    a_src = (float*)take((size_t)N * Hh * 4);
  float*    a_dst = (float*)take((size_t)N * Hh * 4);
  unsigned* amax  = (unsigned*)take((size_t)N * Hh * 4);
  float*    denom = (float*)take((size_t)N * Hh * 4);
  float*    alpha = (float*)take((size_t)EF * Hh * 4);
  int*      s32   = (int*)take((size_t)EF * 4);
  int*      d32   = (int*)take((size_t)EF * 4);
  float*    xh    = (float*)take((size_t)N * HC * 4);
  float*    hA    = (float*)take((size_t)N * HC * 4);
  float*    hB    = (float*)take((size_t)N * HC * 4);

  const int B = 256;
  constexpr int CT = 2;                       // col tiles per wave

  // self-loop attrs (mean of incoming edge_attr per dst)
  hipMemsetAsync(deg, 0, (size_t)N * 4, stream);
  hipMemsetAsync(lsum, 0, (size_t)N * ED * 4, stream);
  k_we<<<1, 64, 0, stream>>>(W_edge, att_edge, we);
  k_deg_loop<<<gridFor(E, B), B, 0, stream>>>(ei, edge_attr, deg, lsum, E);
  k_loop_div<<<gridFor((long)N * ED, B), B, 0, stream>>>(lsum, deg, N * ED);

  const float* hin = x;
  float* houts[Ll] = {hA, hB, hA};
  long rtiles = (N + 15) / 16;
  long gemm_waves = rtiles * (HC / (16 * CT));

  for (int l = 0; l < Ll; ++l) {
    if (l == 0)
      k_gemm_wmma<16, HC, CT><<<gridFor(gemm_waves * 32, B), B, 0, stream>>>(hin, W0, xh, N);
    else
      k_gemm_wmma<HC, HC, CT><<<gridFor(gemm_waves * 32, B), B, 0, stream>>>(
          hin, W12 + (size_t)(l - 1) * HC * HC, xh, N);
    k_asd<<<gridFor((long)N * 32, B), B, 0, stream>>>(xh, att_src + l * HC, att_dst + l * HC,
                                                      a_src, a_dst, N);
    float* outb = houts[l];
    hipMemsetAsync(amax, 0, (size_t)N * Hh * 4, stream);   // 0 < fenc(-inf): safe init
    hipMemsetAsync(denom, 0, (size_t)N * Hh * 4, stream);
    hipMemsetAsync(outb, 0, (size_t)N * HC * 4, stream);
    k_alpha_max<<<gridFor(EF, B), B, 0, stream>>>(ei, edge_attr, lsum, we + l * Hh * ED,
                                                  a_src, a_dst, alpha, amax, s32, d32, E, EF);
    k_exp_denom<<<gridFor(EF, B), B, 0, stream>>>(d32, amax, alpha, denom, EF);
    k_scatter<<<gridFor((long)EF * 32, B), B, 0, stream>>>(s32, d32, alpha, denom, xh, outb, EF);
    k_bias_leaky<<<gridFor((long)N * HC, B), B, 0, stream>>>(outb, bias + l * HC,
                                                             (long)N * HC, HC, 0.01f);
    hin = outb;
  }

  // MLP: hidden = leaky(h @ w1 + b1), out = hidden @ w2 + b2
  long mlp_waves = rtiles * (MH / (16 * CT));
  k_gemm_wmma<HC, MH, CT><<<gridFor(mlp_waves * 32, B), B, 0, stream>>>(hin, mlp_w1, xh, N);
  k_bias_leaky<<<gridFor((long)N * MH, B), B, 0, stream>>>(xh, mlp_b1, (long)N * MH, MH, 0.01f);
  k_mlp2<<<gridFor(N, B), B, 0, stream>>>(xh, mlp_w2, mlp_b2, (float*)d_out, N);
}